// LinearAttentionHead_71433896067705
// MI455X (gfx1250) — compile-verified
//
#include <hip/hip_runtime.h>

typedef float v2f __attribute__((ext_vector_type(2)));
typedef float v8f __attribute__((ext_vector_type(8)));

#define BB 2
#define NN 2048
#define KP 256
#define DD 128
#define TK 64   // k-rows per LDS tile in stage 3

// ---------------------------------------------------------------------------
// Stage 1: Out[m, c] = sum_d X[m, d] * W[c, d] + bias[c]
// One wave per 16x16 output tile, K-loop in steps of 4 via V_WMMA_F32_16X16X4_F32.
// ---------------------------------------------------------------------------
__global__ __launch_bounds__(256) void la_proj_xwT(const float* __restrict__ X,
                                                   const float* __restrict__ W,
                                                   const float* __restrict__ bias,
                                                   float* __restrict__ Out,
                                                   int Mtotal) {
    const int lane  = threadIdx.x & 31;
    const int wave  = blockIdx.x * (blockDim.x >> 5) + (threadIdx.x >> 5);
    const int ctiles = DD / 16;                    // 8
    const int m0 = (wave / ctiles) * 16;
    const int c0 = (wave % ctiles) * 16;
    if (m0 >= Mtotal) return;                      // wave-uniform

    const int l16   = lane & 15;
    const int kbase = (lane >> 4) << 1;            // 0 or 2

    v8f acc = {};
    const float* xrow = X + (m0 + l16) * DD + kbase;   // A[M=l16, K=kbase..kbase+1]
    const float* wrow = W + (c0 + l16) * DD + kbase;   // B[K, N=l16] = W[c, d]

    #pragma unroll
    for (int d = 0; d < DD; d += 4) {
        v2f a = *(const v2f*)(xrow + d);
        v2f b = *(const v2f*)(wrow + d);
        acc = __builtin_amdgcn_wmma_f32_16x16x4_f32(false, a, false, b,
                                                    (short)0, acc, false, false);
    }

    const int col   = c0 + l16;
    const float bv  = bias[col];
    const int rbase = m0 + ((lane >> 4) << 3);     // M = rbase + j
    #pragma unroll
    for (int j = 0; j < 8; ++j)
        Out[(rbase + j) * DD + col] = acc[j] + bv;
}

// ---------------------------------------------------------------------------
// Stage 2: Outp[b, k, c] = sum_n Pm[n, k] * Xn[b, n, c]
// A[m, kk] = Pm[n+kk, k0+m] (transposed read), B[kk, c] = Xn[b, n+kk, c0+c].
// Half-wave b32 loads are lane-coalesced (16 consecutive floats per row).
// ---------------------------------------------------------------------------
__global__ __launch_bounds__(256) void la_lin_proj(const float* __restrict__ Pm,
                                                   const float* __restrict__ Xn,
                                                   float* __restrict__ Outp) {
    const int lane   = threadIdx.x & 31;
    const int wave   = blockIdx.x * (blockDim.x >> 5) + (threadIdx.x >> 5);
    const int ctiles = DD / 16;                    // 8
    const int ktiles = KP / 16;                    // 16
    const int tilesb = ctiles * ktiles;            // 128 tiles per batch
    const int b  = wave / tilesb;
    const int t  = wave % tilesb;
    const int k0 = (t / ctiles) * 16;
    const int c0 = (t % ctiles) * 16;

    const int l16 = lane & 15;
    const int kb  = (lane >> 4) << 1;              // 0 or 2

    v8f acc = {};
    const float* acol = Pm + kb * KP + (k0 + l16);                       // stride KP along n
    const float* brow = Xn + (size_t)b * NN * DD + kb * DD + (c0 + l16); // stride DD along n

    for (int n = 0; n < NN; n += 4) {
        v2f a;
        a.x = acol[n * KP];
        a.y = acol[(n + 1) * KP];
        v2f bvec;
        bvec.x = brow[n * DD];
        bvec.y = brow[(n + 1) * DD];
        acc = __builtin_amdgcn_wmma_f32_16x16x4_f32(false, a, false, bvec,
                                                    (short)0, acc, false, false);
    }

    const int col   = c0 + l16;
    const int rbase = k0 + ((lane >> 4) << 3);
    float* outb = Outp + (size_t)b * KP * DD;
    #pragma unroll
    for (int j = 0; j < 8; ++j)
        outb[(rbase + j) * DD + col] = acc[j];
}

// ---------------------------------------------------------------------------
// Stage 3: per (b, n) row:
//   t[k, c] = QW[n, c] * KWp[k, c] / sqrt(D);  P[k,:] = softmax_c(t[k,:])
//   out[n, c] = sum_k P[k, c] * VWp[k, c]
// 8 waves/block share one batch b, so stage KWp/VWp tiles in LDS via the
// gfx1250 async load-to-LDS path (ASYNCcnt) — 8x less global/L2 traffic.
// ---------------------------------------------------------------------------
__global__ __launch_bounds__(256) void la_attn_lds(const float* __restrict__ QW,   // [B,N,D]
                                                   const float* __restrict__ KWp,  // [B,KP,D]
                                                   const float* __restrict__ VWp,  // [B,KP,D]
                                                   float* __restrict__ Out) {      // [B,N,D]
    __shared__ float sK[TK * DD];   // 32 KB
    __shared__ float sV[TK * DD];   // 32 KB

    const int lane = threadIdx.x & 31;
    const int widx = threadIdx.x >> 5;
    const int row  = blockIdx.x * 8 + widx;        // flat b*NN + n
    const int b    = row >> 11;                    // row / 2048 (NN)

    const float inv_s = 0.08838834764831845f;      // 1/sqrt(128)

    float4 qv = *(const float4*)(QW + (size_t)row * DD + lane * 4);
    qv.x *= inv_s; qv.y *= inv_s; qv.z *= inv_s; qv.w *= inv_s;

    const float* kbase = KWp + (size_t)b * KP * DD;
    const float* vbase = VWp + (size_t)b * KP * DD;

    float4 outv = {0.f, 0.f, 0.f, 0.f};

    for (int k0 = 0; k0 < KP; k0 += TK) {
        // ---- cooperative async copy: 64x128 f32 tile of K and of V ----
        // 256 threads x 8 sweeps x 16B = 32 KB per matrix; source is contiguous.
        #pragma unroll
        for (int s = 0; s < 8; ++s) {
            const int off = threadIdx.x * 4 + s * 1024;           // float idx in tile
            uint32_t dk = (uint32_t)(uintptr_t)(&sK[off]);
            uint32_t dv = (uint32_t)(uintptr_t)(&sV[off]);
            uint64_t gk = (uint64_t)(uintptr_t)(kbase + k0 * DD + off);
            uint64_t gv = (uint64_t)(uintptr_t)(vbase + k0 * DD + off);
            asm volatile("global_load_async_to_lds_b128 %0, %1, off"
                         :: "v"(dk), "v"(gk) : "memory");
            asm volatile("global_load_async_to_lds_b128 %0, %1, off"
                         :: "v"(dv), "v"(gv) : "memory");
        }
        asm volatile("s_wait_asynccnt 0x0" ::: "memory");
        __syncthreads();

        // ---- compute on the LDS tile ----
        #pragma unroll 4
        for (int kk = 0; kk < TK; ++kk) {
            float4 kv = *(const float4*)(&sK[kk * DD + lane * 4]);
            float4 vv = *(const float4*)(&sV[kk * DD + lane * 4]);

            float t0 = qv.x * kv.x;
            float t1 = qv.y * kv.y;
            float t2 = qv.z * kv.z;
            float t3 = qv.w * kv.w;

            float m = fmaxf(fmaxf(t0, t1), fmaxf(t2, t3));
            #pragma unroll
            for (int o = 16; o > 0; o >>= 1)
                m = fmaxf(m, __shfl_xor(m, o, 32));

            float e0 = __expf(t0 - m);
            float e1 = __expf(t1 - m);
            float e2 = __expf(t2 - m);
            float e3 = __expf(t3 - m);

            float z = e0 + e1 + e2 + e3;
            #pragma unroll
            for (int o = 16; o > 0; o >>= 1)
                z += __shfl_xor(z, o, 32);

            float rz = 1.0f / z;
            outv.x += e0 * rz * vv.x;
            outv.y += e1 * rz * vv.y;
            outv.z += e2 * rz * vv.z;
            outv.w += e3 * rz * vv.w;
        }
        __syncthreads();   // protect LDS tile before next async overwrite
    }

    *(float4*)(Out + (size_t)row * DD + lane * 4) = outv;
}

// ---------------------------------------------------------------------------
extern "C" void kernel_launch(void* const* d_in, const int* in_sizes, int n_in,
                              void* d_out, int out_size, void* d_ws, size_t ws_size,
                              hipStream_t stream) {
    (void)in_sizes; (void)n_in; (void)out_size; (void)ws_size;

    const float* Q  = (const float*)d_in[0];
    const float* K  = (const float*)d_in[1];
    const float* V  = (const float*)d_in[2];
    const float* Wq = (const float*)d_in[3];
    const float* bq = (const float*)d_in[4];
    const float* Wk = (const float*)d_in[5];
    const float* bk = (const float*)d_in[6];
    const float* Wv = (const float*)d_in[7];
    const float* bv = (const float*)d_in[8];
    const float* E  = (const float*)d_in[9];
    const float* Fm = (const float*)d_in[10];
    float* out = (float*)d_out;

    // workspace layout (floats): QW | KWn | VWn | KWp | VWp  (~6.5 MB)
    float* ws  = (float*)d_ws;
    float* QW  = ws;
    float* KWn = QW  + (size_t)BB * NN * DD;
    float* VWn = KWn + (size_t)BB * NN * DD;
    float* KWp = VWn + (size_t)BB * NN * DD;
    float* VWp = KWp + (size_t)BB * KP * DD;

    const int Mtotal = BB * NN;                       // 4096 rows
    dim3 blk(256);

    // Stage 1: (4096/16)*(128/16) = 2048 waves = 256 blocks of 8 waves
    la_proj_xwT<<<dim3(256), blk, 0, stream>>>(Q, Wq, bq, QW,  Mtotal);
    la_proj_xwT<<<dim3(256), blk, 0, stream>>>(K, Wk, bk, KWn, Mtotal);
    la_proj_xwT<<<dim3(256), blk, 0, stream>>>(V, Wv, bv, VWn, Mtotal);

    // Stage 2: 2 batches * (256/16)*(128/16) = 256 waves = 32 blocks
    la_lin_proj<<<dim3(32), blk, 0, stream>>>(E,  KWn, KWp);
    la_lin_proj<<<dim3(32), blk, 0, stream>>>(Fm, VWn, VWp);

    // Stage 3: 4096 rows, 8 per block
    la_attn_lds<<<dim3(512), blk, 0, stream>>>(QW, KWp, VWp, out);
}